// DiagMean_81518479278562
// MI455X (gfx1250) — compile-verified
//
#include <hip/hip_runtime.h>

#define TT 2048
#define ND 2049       // number of diagonals (-1024..1024)
#define NGROUPS 65    // ceil(2049/32) diagonal groups of 32
#define MAXNB 8

typedef __attribute__((ext_vector_type(2))) float v2f;
typedef __attribute__((ext_vector_type(8))) float v8f;

// ---------------------------------------------------------------------------
// Kernel 1: streaming diagonal partial sums.
// grid = (65, B), block = 256 (8 waves). Wave w handles row-band w of its
// diagonal group. Lane t owns diagonal d = g*32 + t - 1024 and accumulates in
// a single register. Per row, the wave's 32 lanes read 32 consecutive floats
// -> one coalesced 128B transaction; each matrix element is read exactly once
// (non-temporal: zero reuse, don't thrash the 192MB L2).
// The row loop is split head/mid/tail so the interior is pure load+add.
// ---------------------------------------------------------------------------
__global__ __launch_bounds__(256) void diag_partial_kernel(
    const float* __restrict__ A, float* __restrict__ partial, int NB) {
  const int g    = blockIdx.x;            // diagonal group 0..64
  const int b    = blockIdx.y;            // batch
  const int lane = threadIdx.x & 31;
  const int wv   = threadIdx.x >> 5;      // 0..7 = row band
  if (wv >= NB) return;

  const int didx    = g * 32 + lane;      // 0..2079
  const int d       = didx - 1024;        // signed diagonal offset
  const bool laneok = (didx <= 2048);
  // predicate-only diagonal: invalid lanes get huge d so the column test fails
  const int dpred   = laneok ? d : 0x20000000;

  // union of valid rows for diagonals [d0, d0+31]:
  //   r >= max(0, -(d0+31)),  r <= min(2046, 2046 - d0)
  const int d0 = g * 32 - 1024;
  int rlo = max(0, -(d0 + 31));
  int rhi = min(2047, 2047 - d0);         // exclusive
  if (rhi < rlo) rhi = rlo;
  const int len = rhi - rlo;
  const int per = (len + NB - 1) / NB;
  const int r0  = min(rhi, rlo + wv * per);
  const int r1  = min(rhi, r0 + per);

  // rows where ALL 32 lanes are valid: r in [max(0,-d0), 2016-d0)
  const bool fullgrp = (g < 64);          // g==64 has only lane 0 live
  int aLo = fullgrp ? max(0, -d0)  : 0x3fffffff;
  int aHi = fullgrp ? (2016 - d0)  : 0x3fffffff;  // exclusive
  if (aHi < aLo) aHi = aLo;

  const int hEnd = min(r1, max(r0, aLo)); // head:  [r0,   hEnd) predicated
  const int mEnd = min(r1, max(hEnd, aHi)); // mid: [hEnd, mEnd) unpredicated
                                            // tail:[mEnd, r1)   predicated

  // A[b, r, r+d] = base + r*(TT+1), with base = A + b*TT*TT + d (per lane)
  const float* __restrict__ base = A + (size_t)b * TT * TT + d;

  float acc = 0.0f;
  #pragma unroll 4
  for (int r = r0; r < hEnd; ++r) {
    if ((unsigned)(r + dpred) <= 2046u)
      acc += __builtin_nontemporal_load(base + (size_t)r * (TT + 1));
  }
  #pragma unroll 8
  for (int r = hEnd; r < mEnd; ++r) {     // hot loop: load + add only
    acc += __builtin_nontemporal_load(base + (size_t)r * (TT + 1));
  }
  #pragma unroll 4
  for (int r = mEnd; r < r1; ++r) {
    if ((unsigned)(r + dpred) <= 2046u)
      acc += __builtin_nontemporal_load(base + (size_t)r * (TT + 1));
  }

  if (laneok) {
    partial[((size_t)b * ND + didx) * NB + wv] = acc;
  }
}

// ---------------------------------------------------------------------------
// Kernel 2: finalize. One block (256 threads, 8 waves) per batch.
// Step 1: band-reduce partial[d][0..NB) via V_WMMA_F32_16X16X4_F32:
//   A = 16 diagonals x 4 bands (lanes 0-15: K=0,1; lanes 16-31: K=2,3),
//   B = ones(4x16)  ->  D[m][*] = sum over bands.
// Two chained WMMAs cover 8 bands. Lane 0 / lane 16 extract M=0..7 / 8..15
// per the 16x16 f32 C/D VGPR layout. Out-of-range rows/bands use clamped
// addresses + value selects (no EXEC branching around the WMMAs).
// Step 2: mean per diagonal, block-reduce mean-of-means, write avg - mean.
// ---------------------------------------------------------------------------
__global__ __launch_bounds__(256) void diag_finalize_kernel(
    const float* __restrict__ partial, float* __restrict__ out, int NB) {
  const int b    = blockIdx.x;
  const int tid  = threadIdx.x;
  const int lane = tid & 31;
  const int wv   = tid >> 5;

  __shared__ float S[129 * 16];   // padded diagonal sums (2064)
  __shared__ float Msum[8];

  const int m     = lane & 15;            // A-matrix row within tile
  const int kbase = (lane < 16) ? 0 : 2;  // A-matrix K pair per half-wave

  for (int grp = wv; grp < 129; grp += 8) {
    const int dbase = grp * 16;
    const int drow  = dbase + m;          // diagonal index this lane supplies
    const int  drc   = min(drow, ND - 1); // clamped (always-valid) address
    const bool drok  = (drow < ND);
    const size_t pbase = ((size_t)b * ND + drc) * NB;

    v8f cacc = {};
    v2f ones; ones.x = 1.0f; ones.y = 1.0f;
#if __has_builtin(__builtin_amdgcn_wmma_f32_16x16x4_f32)
    #pragma unroll
    for (int half = 0; half < 2; ++half) {
      const int k0  = kbase + half * 4;
      const int k0c = min(k0,     NB - 1);
      const int k1c = min(k0 + 1, NB - 1);
      float a0 = partial[pbase + k0c];
      float a1 = partial[pbase + k1c];
      if (!drok || k0     >= NB) a0 = 0.0f;   // selects, not branches
      if (!drok || k0 + 1 >= NB) a1 = 0.0f;
      v2f a; a.x = a0; a.y = a1;
      // D = A(16x4) x ones(4x16) + C   (emits v_wmma_f32_16x16x4_f32)
      cacc = __builtin_amdgcn_wmma_f32_16x16x4_f32(
          false, a, false, ones, (short)0, cacc, false, false);
    }
    // C/D layout: VGPR j -> lanes 0-15: M=j ; lanes 16-31: M=8+j (N = lane&15)
    if (lane == 0) {
      #pragma unroll
      for (int j = 0; j < 8; ++j) S[dbase + j] = cacc[j];
    } else if (lane == 16) {
      #pragma unroll
      for (int j = 0; j < 8; ++j) S[dbase + 8 + j] = cacc[j];
    }
#else
    // fallback: scalar band reduction (should not be taken on gfx1250)
    if (lane < 16) {
      float s = 0.0f;
      for (int k = 0; k < NB; ++k) s += drok ? partial[pbase + k] : 0.0f;
      S[dbase + lane] = s;
    }
    (void)cacc; (void)ones;
#endif
  }
  __syncthreads();

  // per-diagonal means + mean-of-means
  float mymean[9];
  float local = 0.0f;
  int cnt = 0;
  for (int d = tid; d < ND; d += 256) {
    const int off = d - 1024;
    const float count = (float)(2047 - (off < 0 ? -off : off));
    const float mr = S[d] / count;
    mymean[cnt++] = mr;
    local += mr;
  }
  #pragma unroll
  for (int s = 16; s > 0; s >>= 1) local += __shfl_xor(local, s, 32);
  if (lane == 0) Msum[wv] = local;
  __syncthreads();
  float tot = 0.0f;
  #pragma unroll
  for (int w = 0; w < 8; ++w) tot += Msum[w];
  const float avg = tot / 2049.0f;

  cnt = 0;
  for (int d = tid; d < ND; d += 256) {
    // centered then negated:  out = -(mean - avg) = avg - mean
    out[(size_t)b * ND + d] = avg - mymean[cnt++];
  }
}

extern "C" void kernel_launch(void* const* d_in, const int* in_sizes, int n_in,
                              void* d_out, int out_size, void* d_ws, size_t ws_size,
                              hipStream_t stream) {
  const float* A   = (const float*)d_in[0];
  float*       out = (float*)d_out;
  float*       ws  = (float*)d_ws;

  const int B = in_sizes[0] / (TT * TT);   // 16 for the reference shapes

  // bands of row-parallelism per diagonal group, limited by workspace
  int NB = MAXNB;
  while (NB > 1 && (size_t)B * ND * NB * sizeof(float) > ws_size) --NB;

  dim3 grid1(NGROUPS, B);
  diag_partial_kernel<<<grid1, 256, 0, stream>>>(A, ws, NB);

  diag_finalize_kernel<<<B, 256, 0, stream>>>(ws, out, NB);
}